// CONV_LSTM_Classifier_73547019976921
// MI455X (gfx1250) — compile-verified
//
#include <hip/hip_runtime.h>
#include <math.h>

#define L      8192
#define NTH    256
#define NWAVE  8
#define N1     64      // L = N1 * N2
#define N2     128
#define TWO_PI 6.283185307179586

typedef __attribute__((ext_vector_type(2))) float v2f;
typedef __attribute__((ext_vector_type(8))) float v8f;

__device__ __forceinline__ v8f wmma4(v2f a, v2f b, v8f c) {
  // D = A(16x4,f32) * B(4x16,f32) + C(16x16,f32)
  return __builtin_amdgcn_wmma_f32_16x16x4_f32(false, a, false, b, (short)0, c,
                                               false, false);
}

__device__ __forceinline__ float blk_sum(float v, volatile float* scr, int tid) {
  for (int off = 16; off >= 1; off >>= 1) v += __shfl_xor(v, off, 32);
  if ((tid & 31) == 0) scr[tid >> 5] = v;
  __syncthreads();
  if (tid == 0) { float r = 0.f; for (int i = 0; i < NWAVE; ++i) r += scr[i]; scr[0] = r; }
  __syncthreads();
  float r = scr[0];
  __syncthreads();
  return r;
}

__device__ __forceinline__ float blk_max(float v, volatile float* scr, int tid) {
  for (int off = 16; off >= 1; off >>= 1) v = fmaxf(v, __shfl_xor(v, off, 32));
  if ((tid & 31) == 0) scr[tid >> 5] = v;
  __syncthreads();
  if (tid == 0) { float r = scr[0]; for (int i = 1; i < NWAVE; ++i) r = fmaxf(r, scr[i]); scr[0] = r; }
  __syncthreads();
  float r = scr[0];
  __syncthreads();
  return r;
}

__device__ __forceinline__ float blk_min(float v, volatile float* scr, int tid) {
  for (int off = 16; off >= 1; off >>= 1) v = fminf(v, __shfl_xor(v, off, 32));
  if ((tid & 31) == 0) scr[tid >> 5] = v;
  __syncthreads();
  if (tid == 0) { float r = scr[0]; for (int i = 1; i < NWAVE; ++i) r = fminf(r, scr[i]); scr[0] = r; }
  __syncthreads();
  float r = scr[0];
  __syncthreads();
  return r;
}

__global__ __launch_bounds__(NTH)
void health_fft_kernel(const float* __restrict__ x, float* __restrict__ out) {
  const int row  = blockIdx.x;
  const int tid  = threadIdx.x;
  const int lane = tid & 31;
  const int wv   = tid >> 5;

  __shared__ float xs[L];        // row data; reused as fr (real part of FFT)
  __shared__ float tw[L];        // tw[j] = cos(2*pi*j/L)
  __shared__ float Zr[L];        // 64x128 intermediate (real)
  __shared__ float Zi[L];        // 64x128 intermediate (imag)
  __shared__ float scr[NWAVE];
  __shared__ int   scri[NWAVE];
  __shared__ float topv[50];
  __shared__ int   topi[50];

  // phase offsets into the cos table (indices are in units of 2*pi/L):
  //   cos(th)  = tw[t]
  //   sin(th)  = cos(th - pi/2) = tw[(t + 6144) & (L-1)]
  //   -sin(th) = cos(th + pi/2) = tw[(t + 2048) & (L-1)]
  const float* src = x + (size_t)row * L;
  float* o = out + (size_t)row * 70;

  // ---- load row (128-bit) + build cos table ----
  {
    const float4* s4 = (const float4*)src;
    float4* x4 = (float4*)xs;
    #pragma unroll
    for (int j = 0; j < L / 4 / NTH; ++j) x4[tid + NTH * j] = s4[tid + NTH * j];
    const float w = (float)(TWO_PI / (double)L);
    for (int i = tid; i < L; i += NTH) tw[i] = cosf(w * (float)i);
  }
  __syncthreads();

  // ================= time-domain stats =================
  // pass 1
  float s = 0.f, s2 = 0.f, sa = 0.f, mxv = -INFINITY, mnv = INFINITY, sd1 = 0.f, sd2 = 0.f;
  for (int i = tid; i < L; i += NTH) {
    float v = xs[i];
    s += v; s2 += v * v; sa += fabsf(v);
    mxv = fmaxf(mxv, v); mnv = fminf(mnv, v);
    if (i >= 1) { float d = v - xs[i - 1]; sd1 += d * d; }
    if (i >= 2) { float dd = v - 2.f * xs[i - 1] + xs[i - 2]; sd2 += dd * dd; }
  }
  const float S   = blk_sum(s,   scr, tid);
  const float S2  = blk_sum(s2,  scr, tid);
  const float SA  = blk_sum(sa,  scr, tid);
  const float MX  = blk_max(mxv, scr, tid);
  const float MN  = blk_min(mnv, scr, tid);
  const float SD1 = blk_sum(sd1, scr, tid);
  const float SD2 = blk_sum(sd2, scr, tid);

  const float mean = S / (float)L;
  const float var  = (S2 - S * S / (float)L) / (float)(L - 1);
  const float stdv = sqrtf(var);
  const float rms  = sqrtf(S2 / (float)L);
  const float sum_d1 = xs[L - 1] - xs[0];                       // telescoping
  const float v1 = (SD1 - sum_d1 * sum_d1 / (float)(L - 1)) / (float)(L - 2);
  const float sum_d2 = (xs[L - 1] - xs[L - 2]) - (xs[1] - xs[0]);
  const float v2 = (SD2 - sum_d2 * sum_d2 / (float)(L - 2)) / (float)(L - 3);

  // pass 2 (needs mean/std)
  float c3 = 0.f, c4 = 0.f, outl = 0.f, zc = 0.f;
  const float thr = 3.0f * stdv;
  for (int i = tid; i < L; i += NTH) {
    float c = xs[i] - mean, cc = c * c;
    c3 += cc * c; c4 += cc * cc;
    if (fabsf(c) > thr) outl += 1.f;
    if (i >= 1) {
      float a = xs[i - 1], b = xs[i];
      int sp = (a > 0.f) - (a < 0.f), sn = (b > 0.f) - (b < 0.f);
      if (sp != sn) zc += 1.f;
    }
  }
  const float C3  = blk_sum(c3,   scr, tid);
  const float C4  = blk_sum(c4,   scr, tid);
  const float OUT = blk_sum(outl, scr, tid);
  const float ZC  = blk_sum(zc,   scr, tid);

  if (tid == 0) {
    float maxabs = fmaxf(fabsf(MX), fabsf(MN));
    o[0]  = mean;
    o[1]  = MX;
    o[2]  = MN;
    o[3]  = MX - MN;
    o[4]  = var;
    o[5]  = rms;
    o[6]  = (C3 / (float)L) / (stdv * stdv * stdv);
    o[7]  = (C4 / (float)L) / (var * var);
    o[8]  = maxabs / rms;
    o[9]  = rms * (float)L / SA;
    o[10] = maxabs * (float)L / SA;
    o[11] = OUT;
    o[12] = ZC / (2.f * (float)L);
    o[13] = var;                  // activity
    o[14] = sqrtf(v1 / var);      // mobility
    o[15] = sqrtf(v2 / v1);       // complexity
  }

  // ================= four-step real-output DFT via f32 WMMA =================
  // n = 128*n1 + n2 ; k = k1 + 64*k2 ; W_L^{nk} = W_64^{n1 k1} W_L^{n2 k1} W_128^{n2 k2}
  // Step1: Y[k1][n2] = sum_n1 (cos - i sin)(2pi n1 k1/64) * xs[128*n1 + n2]
  {
    const int mat  = wv >> 2;                // waves 0-3: Yr (cos), 4-7: Yi (-sin)
    const int ph   = mat ? 2048 : 0;         // -sin(th) == cos(th + pi/2): pure phase shift
    const int half = lane >> 4;
    const int ml   = lane & 15;
    float* dst = mat ? Zi : Zr;
    for (int t = 0; t < 8; ++t) {
      const int tile = (wv & 3) * 8 + t;     // 0..31
      const int mt = tile >> 3, nt = tile & 7;
      const int m  = mt * 16 + ml;           // k1
      const int nc = nt * 16 + ml;           // n2 column
      v8f acc = {};
      #pragma unroll
      for (int kk = 0; kk < 16; ++kk) {
        const int k0 = kk * 4 + half * 2;    // n1
        v2f a, b;
        const int t0 = (m * k0 * (L / N1)) & (L - 1);
        const int t1 = (m * (k0 + 1) * (L / N1)) & (L - 1);
        a.x = tw[(t0 + ph) & (L - 1)];
        a.y = tw[(t1 + ph) & (L - 1)];
        b.x = xs[k0 * N2 + nc];
        b.y = xs[(k0 + 1) * N2 + nc];
        acc = wmma4(a, b, acc);
      }
      #pragma unroll
      for (int r = 0; r < 8; ++r)
        dst[(mt * 16 + r + half * 8) * N2 + nc] = acc[r];
    }
  }
  __syncthreads();

  // twiddle: Z *= (cos - i sin)(2*pi*k1*n2/L), in place
  for (int i = tid; i < L; i += NTH) {
    int k1 = i >> 7, n2 = i & (N2 - 1);
    int t = (k1 * n2) & (L - 1);
    float c = tw[t], sn = tw[(t + 6144) & (L - 1)];
    float yr = Zr[i], yi = Zi[i];
    Zr[i] = yr * c + yi * sn;
    Zi[i] = yi * c - yr * sn;
  }
  __syncthreads();

  // Step2: Re F[k1+64*k2] = sum_n2 Zr*cos(2pi n2 k2/128) + Zi*sin(2pi n2 k2/128)
  float* fr = xs;                            // xs no longer needed -> reuse as fr
  {
    const int half = lane >> 4;
    const int ml   = lane & 15;
    const v2f* Zr2 = (const v2f*)Zr;         // contiguous even pairs -> ds_load_b64
    const v2f* Zi2 = (const v2f*)Zi;
    for (int t = 0; t < 4; ++t) {
      const int tile = wv * 4 + t;           // 0..31
      const int mt = tile >> 3, nt = tile & 7;
      const int m  = mt * 16 + ml;           // k1
      const int nc = nt * 16 + ml;           // k2 column
      const int rowbase2 = (m * N2) >> 1;    // v2f index of row start
      v8f acc = {};
      #pragma unroll 8
      for (int kk = 0; kk < 32; ++kk) {
        const int k0 = kk * 4 + half * 2;    // n2 (even)
        v2f a, b;
        const int t0 = (k0 * nc * (L / N2)) & (L - 1);
        const int t1 = ((k0 + 1) * nc * (L / N2)) & (L - 1);
        a = Zr2[rowbase2 + (k0 >> 1)];
        b.x = tw[t0]; b.y = tw[t1];
        acc = wmma4(a, b, acc);
        a = Zi2[rowbase2 + (k0 >> 1)];
        b.x = tw[(t0 + 6144) & (L - 1)]; b.y = tw[(t1 + 6144) & (L - 1)];
        acc = wmma4(a, b, acc);
      }
      #pragma unroll
      for (int r = 0; r < 8; ++r) {
        int k1 = mt * 16 + r + half * 8;
        fr[k1 + N1 * nc] = acc[r];
      }
    }
  }
  __syncthreads();

  // ================= top-50 of fr (descending, ties -> lower index) =================
  for (int it = 0; it < 50; ++it) {
    float bv = -INFINITY; int bi = L;
    for (int i = tid; i < L; i += NTH) {
      float v = fr[i];
      if (v > bv) { bv = v; bi = i; }        // increasing i => keeps lowest index on ties
    }
    for (int off = 16; off >= 1; off >>= 1) {
      float ov = __shfl_xor(bv, off, 32);
      int   oi = __shfl_xor(bi, off, 32);
      if (ov > bv || (ov == bv && oi < bi)) { bv = ov; bi = oi; }
    }
    if (lane == 0) { scr[wv] = bv; scri[wv] = bi; }
    __syncthreads();
    if (tid == 0) {
      float v = scr[0]; int ix = scri[0];
      for (int w = 1; w < NWAVE; ++w)
        if (scr[w] > v || (scr[w] == v && scri[w] < ix)) { v = scr[w]; ix = scri[w]; }
      topv[it] = v; topi[it] = ix;
      fr[ix] = -INFINITY;
    }
    __syncthreads();
  }

  if (tid == 0) {
    float sidx = 0.f, sv2 = 0.f;
    for (int i = 0; i < 10; ++i) { sidx += (float)topi[i]; sv2 += topv[i] * topv[i]; }
    o[16] = sidx / 10.f;               // top_k_mean_freq
    o[17] = sqrtf(sv2 / 10.f);         // top_k_rms
    o[18] = (float)topi[0];            // max_freq
    o[19] = fabsf(topv[0]);            // max_rms
    for (int i = 0; i < 50; ++i) o[20 + i] = (float)topi[i];
  }
}

extern "C" void kernel_launch(void* const* d_in, const int* in_sizes, int n_in,
                              void* d_out, int out_size, void* d_ws, size_t ws_size,
                              hipStream_t stream) {
  const float* x = (const float*)d_in[0];
  float* out = (float*)d_out;
  const int rows = in_sizes[0] / L;    // 4096
  hipLaunchKernelGGL(health_fft_kernel, dim3(rows), dim3(NTH), 0, stream, x, out);
}